// _ESM_223338300163
// MI455X (gfx1250) — compile-verified
//
#include <hip/hip_runtime.h>
#include <stdint.h>
#include <math.h>

#define NS 2048
#define NT 4096
#define SEAS 7
#define EMB 9
#define SEASC (NT + SEAS)       // 4103 seasonality columns
#define IN 28
#define OUTW 7
#define NW 7

#define SZ_YHAT (NW * NS * IN)  // 401408
#define SZ_WY   (NW * NS * OUTW)// 100352
#define OFF_WY  SZ_YHAT
#define OFF_LEV (OFF_WY + SZ_WY)        // 501760
#define OFF_SEA (OFF_LEV + NS * NT)     // 8890368

#define CHUNK 112   // recurrence steps per LDS chunk (multiple of 28)
#define ROWF  116   // floats fetched per row per full chunk (29 x 16B)
#define LANES 32
#define NFULL 36    // chunks 0..35 are full; chunk 36 is the 64-float tail

// ---- CDNA5 async global->LDS copy (ASYNCcnt path), raw ISA form ----
// VDST = per-lane LDS byte address VGPR, VADDR = 64-bit global addr pair.
// INST_OFFSET is added to BOTH the LDS and the global address (ISA §10.7),
// so one base address pair serves the whole chunk.
template <int OFF>
__device__ __forceinline__ void async_copy16(const float* g, float* l) {
  asm volatile("global_load_async_to_lds_b128 %0, %1, off offset:%c2"
               :: "v"((unsigned)(uintptr_t)l),
                  "v"((unsigned long long)(uintptr_t)g),
                  "i"(OFF));
}

template <int I, int N>
struct AsyncRun {
  static __device__ __forceinline__ void run(const float* g, float* l) {
    async_copy16<I * 16>(g, l);
    AsyncRun<I + 1, N>::run(g, l);
  }
};
template <int N>
struct AsyncRun<N, N> {
  static __device__ __forceinline__ void run(const float*, float*) {}
};

__device__ __forceinline__ void prefetch_full(const float* g, float* l) {
  AsyncRun<0, 29>::run(g, l);   // 29 x 16B = 116 floats per lane-row
}
__device__ __forceinline__ void prefetch_last(const float* g, float* l) {
  AsyncRun<0, 16>::run(g, l);   // 16 x 16B = 64 floats per lane-row
}

#define WAIT_ASYNC(n) asm volatile("s_wait_asynccnt " #n ::: "memory")

// fast reciprocal: v_rcp_f32 + 1 Newton step (~1 ulp; all operands are
// well-scaled positives here, no denorm risk)
__device__ __forceinline__ float frcp(float x) {
  float r = __builtin_amdgcn_rcpf(x);
  return r * __builtin_fmaf(-x, r, 2.0f);
}

// One smoothing step. p = (t-1) % 7 is a literal so the season ring and its
// reciprocal shadow stay in named registers.  The level recurrence is exactly
// one FMA deep per step: nl = oml*level + (lev_sms*yt*rr), where the addend
// uses rcp(season) computed 7 steps earlier (off the critical path).
#define STEP(p, j)                                                     \
  do {                                                                 \
    float yt = ypl[j];                                                 \
    float nl = __builtin_fmaf(oml, level, lev_sms * (yt * rr##p));     \
    float rl = frcp(nl);                                               \
    float ns = __builtin_fmaf(oms, r##p, seas_sms * (yt * rl));        \
    r##p = ns;                                                         \
    rr##p = frcp(ns);                                                  \
    level = nl;                                                        \
    levp[tb + (j)] = nl;                                               \
    seap[tb + (j) + SEAS] = ns;                                        \
  } while (0)

#define BLK28()                                                          \
  do {                                                                   \
    STEP(0, 0);  STEP(1, 1);  STEP(2, 2);  STEP(3, 3);  STEP(4, 4);      \
    STEP(5, 5);  STEP(6, 6);  STEP(0, 7);  STEP(1, 8);  STEP(2, 9);      \
    STEP(3, 10); STEP(4, 11); STEP(5, 12); STEP(6, 13); STEP(0, 14);     \
    STEP(1, 15); STEP(2, 16); STEP(3, 17); STEP(4, 18); STEP(5, 19);     \
    STEP(6, 20); STEP(0, 21); STEP(1, 22); STEP(2, 23); STEP(3, 24);     \
    STEP(4, 25); STEP(5, 26); STEP(6, 27);                               \
  } while (0)

#define CHUNK4(TB)                                                        \
  do {                                                                    \
    { const float* ypl = lbase + 1;  const int tb = (TB) + 1;  BLK28(); } \
    { const float* ypl = lbase + 29; const int tb = (TB) + 29; BLK28(); } \
    { const float* ypl = lbase + 57; const int tb = (TB) + 57; BLK28(); } \
    { const float* ypl = lbase + 85; const int tb = (TB) + 85; BLK28(); } \
  } while (0)

// One wave32 per block: the wave owns its ASYNCcnt, no barriers needed.
__global__ __launch_bounds__(LANES) void esrnn_scan(
    const float* __restrict__ y, const int* __restrict__ idxs,
    const float* __restrict__ ew, float* __restrict__ out) {
  __shared__ __align__(16) float buf[2][LANES][ROWF];  // double buffer, ~29 KB
  const int lane = threadIdx.x;
  const int s = blockIdx.x * LANES + lane;  // one series per thread
  const float* grow = y + (size_t)s * NT;
  float* levp = out + OFF_LEV + (size_t)s * NT;
  float* seap = out + OFF_SEA + (size_t)s * SEASC;

  // kick off the async pipeline before the scalar prologue math
  prefetch_full(grow, &buf[0][lane][0]);
  prefetch_full(grow + CHUNK, &buf[1][lane][0]);

  const float* e = ew + (size_t)idxs[s] * EMB;
  const float lev_sms  = 1.0f / (1.0f + expf(-e[0]));
  const float seas_sms = 1.0f / (1.0f + expf(-e[1]));
  const float oml = 1.0f - lev_sms, oms = 1.0f - seas_sms;
  const float is0 = expf(e[2]), is1 = expf(e[3]), is2 = expf(e[4]),
              is3 = expf(e[5]), is4 = expf(e[6]), is5 = expf(e[7]),
              is6 = expf(e[8]);

  float level = grow[0] / is0;  // level0 (precise divide, once)
  levp[0] = level;
  seap[0] = is0; seap[1] = is1; seap[2] = is2; seap[3] = is3;
  seap[4] = is4; seap[5] = is5; seap[6] = is6; seap[7] = is0;

  // season ring = rotate(init_seas, 1); shadow reciprocals
  float r0 = is1, r1 = is2, r2 = is3, r3 = is4, r4 = is5, r5 = is6, r6 = is0;
  float rr0 = frcp(r0), rr1 = frcp(r1), rr2 = frcp(r2), rr3 = frcp(r3),
        rr4 = frcp(r4), rr5 = frcp(r5), rr6 = frcp(r6);

  // chunks 0..34: wait for chunk c (next full chunk c+1 still in flight: 29
  // ops), issue chunk c+2, process 4x28 steps.
  for (int c = 0; c < NFULL - 1; ++c) {
    WAIT_ASYNC(29);
    const int n = c + 2;
    float* nlp = &buf[n & 1][lane][0];
    if (n < NFULL) prefetch_full(grow + (size_t)n * CHUNK, nlp);
    else           prefetch_last(grow + (size_t)NFULL * CHUNK, nlp);
    const float* lbase = &buf[c & 1][lane][0];
    CHUNK4(c * CHUNK);
  }
  {  // chunk 35 (tail chunk 36 still in flight: 16 ops)
    WAIT_ASYNC(16);
    const float* lbase = &buf[(NFULL - 1) & 1][lane][0];
    CHUNK4((NFULL - 1) * CHUNK);
  }
  {  // chunk 36: t = 4033..4095 (2x28 + 7 steps)
    WAIT_ASYNC(0);
    const float* lbase = &buf[NFULL & 1][lane][0];
    { const float* ypl = lbase + 1;  const int tb = NFULL * CHUNK + 1;  BLK28(); }
    { const float* ypl = lbase + 29; const int tb = NFULL * CHUNK + 29; BLK28(); }
    { const float* ypl = lbase + 57; const int tb = NFULL * CHUNK + 57;
      STEP(0, 0); STEP(1, 1); STEP(2, 2); STEP(3, 3); STEP(4, 4);
      STEP(5, 5); STEP(6, 6); }
  }
}

// windows_y_hat[w,s,i] = log(y[s,4062+w+i] / levels[s,4089+w] / seas[s,4062+w+i])
// plus zero-fill of windows_y (harness poisons d_out).
__global__ void esrnn_windows(const float* __restrict__ y, float* out) {
  const int idx = blockIdx.x * blockDim.x + threadIdx.x;
  if (idx < SZ_WY) out[OFF_WY + idx] = 0.0f;
  if (idx < SZ_YHAT) {
    const int w = idx / (NS * IN);
    const int r = idx - w * (NS * IN);
    const int s = r / IN;
    const int i = r - s * IN;
    const int t = (NT - IN - OUTW + 1) + w + i;           // 4062 + w + i
    const float yv = y[(size_t)s * NT + t];
    const float lv = out[OFF_LEV + (size_t)s * NT + (NT - OUTW + w)];  // 4089+w
    const float sv = out[OFF_SEA + (size_t)s * SEASC + t];
    out[idx] = logf(yv / lv / sv);
  }
}

extern "C" void kernel_launch(void* const* d_in, const int* in_sizes, int n_in,
                              void* d_out, int out_size, void* d_ws, size_t ws_size,
                              hipStream_t stream) {
  (void)in_sizes; (void)n_in; (void)out_size; (void)d_ws; (void)ws_size;
  const float* y    = (const float*)d_in[0];
  const int*   idxs = (const int*)d_in[1];
  const float* ew   = (const float*)d_in[2];
  float* out = (float*)d_out;
  esrnn_scan<<<NS / LANES, LANES, 0, stream>>>(y, idxs, ew, out);
  esrnn_windows<<<(SZ_YHAT + 255) / 256, 256, 0, stream>>>(y, out);
}